// Net_3393024164211
// MI455X (gfx1250) — compile-verified
//
#include <hip/hip_runtime.h>
#include <hip/hip_bf16.h>
#include <stdint.h>

#define NN 50000
#define NE 1600000
// KSIZE=5 -> idx clipped to [0,3], strides {1,5,25}, K=125 rows of 8 floats.

typedef float v2f __attribute__((ext_vector_type(2)));
typedef float v8f __attribute__((ext_vector_type(8)));
typedef unsigned int v4u __attribute__((ext_vector_type(4)));
typedef int v4i __attribute__((ext_vector_type(4)));
typedef int v8i __attribute__((ext_vector_type(8)));

__device__ __forceinline__ float elu1(float v) {
  return v > 0.0f ? v : expm1f(v);
}

// The TDM writes LDS behind the compiler's back; escape the LDS pointer and
// clobber memory so clang cannot fold reads of the "never-written" shared
// array to poison (which it did in round 1: ds count collapsed to 1).
__device__ __forceinline__ void lds_publish(const void* p) {
  asm volatile("" : : "v"(p) : "memory");
}

// Stage nfloats f32 from global g into LDS offset 0 via the Tensor Data Mover.
// 1D tensor/tile: tensor_dim0 = tile_dim0 = nfloats, data_size = 4B.
__device__ __forceinline__ void tdm_stage_to_lds(const float* g, int nfloats) {
  uint64_t ga = (uint64_t)(uintptr_t)g;
  v4u g0;
  g0.x = 1u;                                   // count=1 (valid user descriptor)
  g0.y = 0u;                                   // lds_addr = 0 (bytes)
  g0.z = (unsigned)(ga & 0xFFFFFFFFu);         // global_addr[31:0]
  g0.w = (unsigned)((ga >> 32) & 0x1FFFFFFu)   // global_addr[56:32]
         | (2u << 30);                         // type = 2 ("image")
  unsigned nf = (unsigned)nfloats;
  v8i g1;
  g1[0] = 0x00020000;                          // data_size=2 (4 bytes); mask=0
  g1[1] = (int)((nf & 0xFFFFu) << 16);         // tensor_dim0[15:0] @ [31:16]
  g1[2] = (int)((nf >> 16) | (1u << 16));      // tensor_dim0[31:16] | tensor_dim1=1
  g1[3] = (int)((nf & 0xFFFFu) << 16);         // tensor_dim1 hi=0 | tile_dim0
  g1[4] = (int)(1u | (1u << 16));              // tile_dim1=1 | tile_dim2=1
  g1[5] = (int)nf;                             // tensor_dim0_stride[31:0]
  g1[6] = (int)((nf & 0xFFFFu) << 16);         // stride0 hi=0 | stride1[15:0]
  g1[7] = (int)(nf >> 16);                     // stride1[47:16]
  v4i gz = {0, 0, 0, 0};
#if __clang_major__ >= 23
  v8i gz8 = {0, 0, 0, 0, 0, 0, 0, 0};
  __builtin_amdgcn_tensor_load_to_lds(g0, g1, gz, gz, gz8, 0);
#else
  __builtin_amdgcn_tensor_load_to_lds(g0, g1, gz, gz, 0);
#endif
  __builtin_amdgcn_s_wait_tensorcnt(0);
}

__global__ __launch_bounds__(256) void zero_kernel(float* __restrict__ p, int n) {
  int i = blockIdx.x * blockDim.x + threadIdx.x;
  if (i < n) p[i] = 0.0f;
}

// Layer-1 edge kernel: msg[e,c] = x[src] * sum_b basis_b * W1[kidx_b, 0, c]
__global__ __launch_bounds__(256) void edge1_kernel(
    const float* __restrict__ x, const long long* __restrict__ ei,
    const float* __restrict__ pseudo, const float* __restrict__ W1,
    float* __restrict__ acc1) {
  __shared__ float sW[1000];  // 125 x 8
  if ((threadIdx.x >> 5) == 0) tdm_stage_to_lds(W1, 1000);
  __syncthreads();
  lds_publish(sW);
  int e = blockIdx.x * blockDim.x + threadIdx.x;
  if (e >= NE) return;
  int s = (int)ei[e];
  int d = (int)ei[NE + e];
  float p0 = pseudo[3 * e + 0], p1 = pseudo[3 * e + 1], p2 = pseudo[3 * e + 2];
  float q0 = p0 * 4.0f, q1 = p1 * 4.0f, q2 = p2 * 4.0f;
  float g0 = fminf(fmaxf(floorf(q0), 0.0f), 3.0f);
  float g1 = fminf(fmaxf(floorf(q1), 0.0f), 3.0f);
  float g2 = fminf(fmaxf(floorf(q2), 0.0f), 3.0f);
  int i0 = (int)g0, i1 = (int)g1, i2 = (int)g2;
  float r0 = q0 - g0, r1 = q1 - g1, r2 = q2 - g2;
  int kbase = i0 + 5 * i1 + 25 * i2;
  float w[8];
#pragma unroll
  for (int c = 0; c < 8; ++c) w[c] = 0.0f;
#pragma unroll
  for (int b = 0; b < 8; ++b) {
    int k = kbase + (b & 1) + 5 * ((b >> 1) & 1) + 25 * ((b >> 2) & 1);
    float basis = ((b & 1) ? r0 : 1.0f - r0) *
                  ((b & 2) ? r1 : 1.0f - r1) *
                  ((b & 4) ? r2 : 1.0f - r2);
    const float* wr = &sW[k * 8];
#pragma unroll
    for (int c = 0; c < 8; ++c) w[c] = fmaf(basis, wr[c], w[c]);
  }
  float xj = x[s];
  float* a = acc1 + (size_t)d * 8;
#pragma unroll
  for (int c = 0; c < 8; ++c) atomicAdd(a + c, xj * w[c]);
}

// Node update after layer 1: h = elu(acc1 + x*root1 + bias1)
__global__ __launch_bounds__(256) void node1_kernel(
    const float* __restrict__ x, const float* __restrict__ root1,
    const float* __restrict__ bias1, const float* __restrict__ acc1,
    float* __restrict__ h) {
  int n = blockIdx.x * blockDim.x + threadIdx.x;
  if (n >= NN) return;
  float xv = x[n];
  const float4* a4 = (const float4*)(acc1 + (size_t)n * 8);
  float4 a0 = a4[0], a1 = a4[1];
  const float4* r4 = (const float4*)root1;
  float4 r0 = r4[0], r1 = r4[1];
  const float4* b4 = (const float4*)bias1;
  float4 b0 = b4[0], b1 = b4[1];
  float4 o0, o1;
  o0.x = elu1(fmaf(xv, r0.x, a0.x) + b0.x);
  o0.y = elu1(fmaf(xv, r0.y, a0.y) + b0.y);
  o0.z = elu1(fmaf(xv, r0.z, a0.z) + b0.z);
  o0.w = elu1(fmaf(xv, r0.w, a0.w) + b0.w);
  o1.x = elu1(fmaf(xv, r1.x, a1.x) + b1.x);
  o1.y = elu1(fmaf(xv, r1.y, a1.y) + b1.y);
  o1.z = elu1(fmaf(xv, r1.z, a1.z) + b1.z);
  o1.w = elu1(fmaf(xv, r1.w, a1.w) + b1.w);
  float4* h4 = (float4*)(h + (size_t)n * 8);
  h4[0] = o0;
  h4[1] = o1;
}

// Layer-2 edge kernel: s = dot(h[src], sum_b basis_b * W2[kidx_b,:,0])
__global__ __launch_bounds__(256) void edge2_kernel(
    const float* __restrict__ h, const long long* __restrict__ ei,
    const float* __restrict__ pseudo, const float* __restrict__ W2,
    float* __restrict__ acc2) {
  __shared__ float sW[1000];  // 125 x 8
  if ((threadIdx.x >> 5) == 0) tdm_stage_to_lds(W2, 1000);
  __syncthreads();
  lds_publish(sW);
  int e = blockIdx.x * blockDim.x + threadIdx.x;
  if (e >= NE) return;
  int s = (int)ei[e];
  int d = (int)ei[NE + e];
  float p0 = pseudo[3 * e + 0], p1 = pseudo[3 * e + 1], p2 = pseudo[3 * e + 2];
  float q0 = p0 * 4.0f, q1 = p1 * 4.0f, q2 = p2 * 4.0f;
  float g0 = fminf(fmaxf(floorf(q0), 0.0f), 3.0f);
  float g1 = fminf(fmaxf(floorf(q1), 0.0f), 3.0f);
  float g2 = fminf(fmaxf(floorf(q2), 0.0f), 3.0f);
  int i0 = (int)g0, i1 = (int)g1, i2 = (int)g2;
  float r0 = q0 - g0, r1 = q1 - g1, r2 = q2 - g2;
  int kbase = i0 + 5 * i1 + 25 * i2;
  float w[8];
#pragma unroll
  for (int c = 0; c < 8; ++c) w[c] = 0.0f;
#pragma unroll
  for (int b = 0; b < 8; ++b) {
    int k = kbase + (b & 1) + 5 * ((b >> 1) & 1) + 25 * ((b >> 2) & 1);
    float basis = ((b & 1) ? r0 : 1.0f - r0) *
                  ((b & 2) ? r1 : 1.0f - r1) *
                  ((b & 4) ? r2 : 1.0f - r2);
    const float* wr = &sW[k * 8];
#pragma unroll
    for (int c = 0; c < 8; ++c) w[c] = fmaf(basis, wr[c], w[c]);
  }
  const float4* h4 = (const float4*)(h + (size_t)s * 8);
  float4 h0 = h4[0], h1 = h4[1];
  float acc = h0.x * w[0] + h0.y * w[1] + h0.z * w[2] + h0.w * w[3] +
              h1.x * w[4] + h1.y * w[5] + h1.z * w[6] + h1.w * w[7];
  atomicAdd(acc2 + d, acc);
}

// Final node update via WMMA: out = elu(acc2 + h @ root2 + bias2)
// One wave handles a 16-node tile: A = h_tile (16x8 f32, two 16x4 WMMAs),
// B = root2 broadcast into all 16 columns -> every column of D is the GEMV.
// D layout (16x16 f32): VGPR r, lanes 0-15 = row r, lanes 16-31 = row 8+r.
__global__ __launch_bounds__(256) void node2_kernel(
    const float* __restrict__ h, const float* __restrict__ root2,
    const float* __restrict__ bias2, const float* __restrict__ acc2,
    float* __restrict__ out) {
  int wave = threadIdx.x >> 5;
  int lane = threadIdx.x & 31;
  int base = (blockIdx.x * 8 + wave) * 16;
  if (base >= NN) return;  // wave-uniform; active waves keep EXEC all-1s
  int m = lane & 15;       // row (node within tile)
  int kh = lane >> 4;      // K-half select per documented 16x4 f32 A layout
  const float* hp = h + (size_t)(base + m) * 8;
  v2f a0 = {hp[2 * kh + 0], hp[2 * kh + 1]};          // K = 0..3
  v2f a1 = {hp[4 + 2 * kh + 0], hp[4 + 2 * kh + 1]};  // K = 4..7
  v2f b0 = {root2[2 * kh + 0], root2[2 * kh + 1]};    // broadcast to all N
  v2f b1 = {root2[4 + 2 * kh + 0], root2[4 + 2 * kh + 1]};
  v8f c = {0.f, 0.f, 0.f, 0.f, 0.f, 0.f, 0.f, 0.f};
  c = __builtin_amdgcn_wmma_f32_16x16x4_f32(false, a0, false, b0, (short)0, c,
                                            false, false);
  c = __builtin_amdgcn_wmma_f32_16x16x4_f32(false, a1, false, b1, (short)0, c,
                                            false, false);
  // All columns identical; pick VGPR (m&7), rows 8-15 live in the upper half.
  int mm = m & 7;
  float sel = c[0];
  sel = (mm == 1) ? c[1] : sel;
  sel = (mm == 2) ? c[2] : sel;
  sel = (mm == 3) ? c[3] : sel;
  sel = (mm == 4) ? c[4] : sel;
  sel = (mm == 5) ? c[5] : sel;
  sel = (mm == 6) ? c[6] : sel;
  sel = (mm == 7) ? c[7] : sel;
  float other = __shfl_xor(sel, 16, 32);  // fetch rows 8..15 from upper half
  float res = (m < 8) ? sel : other;
  if (lane < 16) {
    int n = base + m;
    out[n] = elu1(res + acc2[n] + bias2[0]);
  }
}

extern "C" void kernel_launch(void* const* d_in, const int* in_sizes, int n_in,
                              void* d_out, int out_size, void* d_ws,
                              size_t ws_size, hipStream_t stream) {
  const float* x = (const float*)d_in[0];
  const long long* ei = (const long long*)d_in[1];  // [2, E] int64
  const float* pseudo = (const float*)d_in[2];      // [E, 3]
  const float* W1 = (const float*)d_in[3];          // [125,1,8]
  const float* root1 = (const float*)d_in[4];       // [1,8]
  const float* bias1 = (const float*)d_in[5];       // [8]
  const float* W2 = (const float*)d_in[6];          // [125,8,1]
  const float* root2 = (const float*)d_in[7];       // [8,1]
  const float* bias2 = (const float*)d_in[8];       // [1]
  float* out = (float*)d_out;

  float* acc1 = (float*)d_ws;           // NN*8 floats
  float* hbuf = acc1 + (size_t)NN * 8;  // NN*8 floats
  float* acc2 = hbuf + (size_t)NN * 8;  // NN floats

  zero_kernel<<<(NN * 8 + 255) / 256, 256, 0, stream>>>(acc1, NN * 8);
  zero_kernel<<<(NN + 255) / 256, 256, 0, stream>>>(acc2, NN);
  edge1_kernel<<<(NE + 255) / 256, 256, 0, stream>>>(x, ei, pseudo, W1, acc1);
  node1_kernel<<<(NN + 255) / 256, 256, 0, stream>>>(x, root1, bias1, acc1,
                                                     hbuf);
  edge2_kernel<<<(NE + 255) / 256, 256, 0, stream>>>(hbuf, ei, pseudo, W2,
                                                     acc2);
  node2_kernel<<<(NN + 127) / 128, 256, 0, stream>>>(hbuf, root2, bias2, acc2,
                                                     out);
}